// HippoSSKernel_7421703488211
// MI455X (gfx1250) — compile-verified
//
#include <hip/hip_runtime.h>
#include <math.h>

// Problem constants from the reference (H, N_FULL, L_KERNEL, CHANNELS, RANK) =
// (256, 128, 2048, 1, 1); after conjugate pairing Nh = 64, Lh = L/2+1 = 1025.
#define HH 256
#define NH 64
#define LL 2048
#define LH 1025
#define SEG 1028   // each of Re / Im segment padded to multiple of 4
#define KP  2056   // GEMM K dimension: [Re(1025)+pad3 | Im(1025)+pad3]

typedef __attribute__((ext_vector_type(2))) float v2f;
typedef __attribute__((ext_vector_type(8))) float v8f;

__device__ __forceinline__ float2 cmul(float2 a, float2 b) {
    return make_float2(a.x * b.x - a.y * b.y, a.x * b.y + a.y * b.x);
}

// ---------------------------------------------------------------------------
// Stage 1: Cauchy kernel + rank-1 Woodbury correction at the 1025 bilinear
// frequencies. Writes the pre-weighted real GEMM "A" matrix (HH x KP) into ws:
//   A[h, l]        =  w_l * Re(k_f[h,l])     (w_0 = w_1024 = 1/L, else 2/L)
//   A[h, SEG + l]  = -w_l * Im(k_f[h,l])     (sign of the -sin basis folded in;
//                                             Im at DC / Nyquist zeroed,
//                                             matching irfft semantics)
//   pad columns (1025..1027, 2053..2055)     = 0
// z_l = 2(1-w)/(1+w) = i * 2 tan(pi l / L) is purely imaginary, so the
// conjugate-pole half of the sum is conj(v / (-z - w_dt)).
// 2/(1+omega) simplifies to 1 + i*tan(theta/2). Nyquist handled via its
// analytic limit: k_f -> dt * sum_n Re(B_n C_n).
// ---------------------------------------------------------------------------
__global__ void hippo_cauchy_kernel(const float* __restrict__ w_ri,
                                    const float* __restrict__ P_ri,
                                    const float* __restrict__ B_ri,
                                    const float* __restrict__ C_ri,
                                    const float* __restrict__ log_dt,
                                    float* __restrict__ Amat)
{
    int idx = blockIdx.x * blockDim.x + threadIdx.x;
    if (idx >= HH * LH) return;
    int h = idx / LH;
    int l = idx - h * LH;

    const float* wr = w_ri + (size_t)h * NH * 2;
    const float* pr = P_ri + (size_t)h * NH * 2;   // (RANK=1, H, NH, 2)
    const float* br = B_ri + (size_t)h * NH * 2;   // (1, H, NH, 2)
    const float* cr = C_ri + (size_t)h * NH * 2;   // (CHANNELS=1, H, NH, 2)

    float dt = expf(log_dt[h]);
    float* arow = Amat + (size_t)h * KP;

    if (l == LH - 1) {
        // Nyquist bin: lim_{omega->-1} of the full expression = dt*sum Re(B*C)
        float s = 0.f;
        for (int n = 0; n < NH; ++n)
            s += br[2*n] * cr[2*n] - br[2*n+1] * cr[2*n+1];
        arow[l]       = (dt * s) * (1.0f / (float)LL);
        arow[SEG + l] = 0.f;   // irfft ignores Im at Nyquist
        return;
    }

    const float theta = (float)l * 3.0679615757712823e-3f;  // 2*pi/2048
    const float zy = 2.0f * tanf(0.5f * theta);             // z = i*zy

    float2 r00 = {0.f,0.f}, r01 = {0.f,0.f}, r10 = {0.f,0.f}, r11 = {0.f,0.f};
    for (int n = 0; n < NH; ++n) {
        float2 wv = make_float2(wr[2*n] * dt, wr[2*n+1] * dt);   // w_dt
        float2 Bv = make_float2(br[2*n], br[2*n+1]);
        float2 Pv = make_float2(pr[2*n], pr[2*n+1]);
        float2 Cv = make_float2(cr[2*n], cr[2*n+1]);
        float2 Pc = make_float2(Pv.x, -Pv.y);
        float2 v00 = cmul(Bv, Cv);   // Bcat0 * Ccat0
        float2 v01 = cmul(Bv, Pc);   // Bcat0 * Ccat1 = B * conj(P)
        float2 v10 = cmul(Pv, Cv);   // Bcat1 * Ccat0
        float2 v11 = cmul(Pv, Pc);   // Bcat1 * Ccat1 = |P|^2

        float d1x = -wv.x, d1y = zy - wv.y;          // z - w_dt
        float is1 = 1.0f / (d1x*d1x + d1y*d1y);
        float2 inv1 = make_float2(d1x * is1, -d1y * is1);
        float d2x = -wv.x, d2y = -zy - wv.y;         // -z - w_dt
        float is2 = 1.0f / (d2x*d2x + d2y*d2y);
        float2 inv2 = make_float2(d2x * is2, -d2y * is2);

        // r += v*inv1 + conj(v*inv2)
        float2 t1, t2;
        t1 = cmul(v00, inv1); t2 = cmul(v00, inv2); r00.x += t1.x + t2.x; r00.y += t1.y - t2.y;
        t1 = cmul(v01, inv1); t2 = cmul(v01, inv2); r01.x += t1.x + t2.x; r01.y += t1.y - t2.y;
        t1 = cmul(v10, inv1); t2 = cmul(v10, inv2); r10.x += t1.x + t2.x; r10.y += t1.y - t2.y;
        t1 = cmul(v11, inv1); t2 = cmul(v11, inv2); r11.x += t1.x + t2.x; r11.y += t1.y - t2.y;
    }
    r00.x *= dt; r00.y *= dt;  r01.x *= dt; r01.y *= dt;
    r10.x *= dt; r10.y *= dt;  r11.x *= dt; r11.y *= dt;

    // k_f = r00 - r01*r10/(1+r11)
    float px = 1.f + r11.x, py = r11.y;
    float ps = 1.0f / (px*px + py*py);
    float2 ipn  = make_float2(px * ps, -py * ps);
    float2 corr = cmul(cmul(r01, r10), ipn);
    float2 kf   = make_float2(r00.x - corr.x, r00.y - corr.y);
    // k_f *= 2/(1+omega) = 1 + i*tan(theta/2)
    float ci = 0.5f * zy;
    float2 kf2 = make_float2(kf.x - kf.y * ci, kf.y + kf.x * ci);

    float wgt = (l == 0) ? (1.0f / (float)LL) : (2.0f / (float)LL);
    arow[l]       = kf2.x * wgt;
    arow[SEG + l] = (l == 0) ? 0.f : -kf2.y * wgt;  // -sin sign folded in
    if (l == 0) {
        arow[1025] = 0.f; arow[1026] = 0.f; arow[1027] = 0.f;   // Re pad
        arow[2053] = 0.f; arow[2054] = 0.f; arow[2055] = 0.f;   // Im pad
    }
}

// ---------------------------------------------------------------------------
// Stage 2: irfft as a dense fp32 GEMM on the WMMA pipe.
//   K[h,t] = sum_k A[h,k] * Bm[k,t]
//   Bm[l, t]       = cos(2*pi*l*t/L)                    l in [0,1028)
//   Bm[SEG + l, t] = sin(2*pi*l*t/L) = cos(2*pi*(l*t - 512)/L)
// (the -sin sign lives in A; pad rows multiply A==0, so their basis value is
//  irrelevant -> no bounds checks). Basis comes from an 8 KB LDS cosine table
// with exact integer phase reduction; the phase advances incrementally by
// (4*t) mod 2048 per K-step, so the hot loop is branch-free and lane-uniform:
//   1x global_load_b64 (A frag) + 2x ds_load_b32 (basis) + 1x wmma.
// 8 waves per block share the table; each wave owns one 16x16 output tile.
// EXEC is all-ones at every WMMA. Fragment layouts per CDNA5 ISA 32-bit
// tables:
//   A 16x4 : lanes 0-15 -> (M=lane, K=k0+{0,1}), lanes 16-31 -> K=k0+{2,3}
//   B 4x16 : lanes 0-15 -> (N=lane, K=k0+{0,1}), lanes 16-31 -> K=k0+{2,3}
//   C 16x16: vgpr r, lanes 0-15 -> (M=r, N=lane); lanes 16-31 -> (M=r+8, N=lane-16)
// ---------------------------------------------------------------------------
__global__ __launch_bounds__(256)
void hippo_idft_wmma_kernel(const float* __restrict__ Amat,
                            float* __restrict__ out)
{
    __shared__ float costab[LL];                  // 8 KB of 320 KB WGP LDS
    const int tid = threadIdx.x;
    for (int i = tid; i < LL; i += 256)
        costab[i] = cosf((float)i * 3.0679615757712823e-3f);  // 2*pi/2048
    __syncthreads();

    const int wave = tid >> 5;
    const int lane = tid & 31;
    const int m0 = blockIdx.y * 16;                       // head tile
    const int n0 = (blockIdx.x * 8 + wave) * 16;          // time tile
    const int half = lane >> 4;                           // 0: K={0,1}, 1: K={2,3}
    const int lm = lane & 15;
    const int t = n0 + lm;

    const float* Arow = Amat + (size_t)(m0 + lm) * KP + 2 * half;
    const int step = (4 * t) & (LL - 1);                  // phase advance per K-step
    const int l0 = 2 * half;

    v8f c = {};

    // ---- segment 1: cosine rows [0, 1028) ----
    int p = (l0 * t) & (LL - 1);
    for (int k0 = 0; k0 < SEG; k0 += 4) {
        v2f a = *(const v2f*)(Arow + k0);                 // A[m, k0+l0 .. +1]
        v2f b;
        b.x = costab[p];
        b.y = costab[(p + t) & (LL - 1)];
        c = __builtin_amdgcn_wmma_f32_16x16x4_f32(
                false, a, false, b, (short)0, c, false, false);
        p = (p + step) & (LL - 1);
    }

    // ---- segment 2: sine rows [SEG, 2*SEG): sin x = cos(x - pi/2) ----
    p = (l0 * t + 1536) & (LL - 1);                       // -512 == +1536 (mod 2048)
    const float* Arow2 = Arow + SEG;
    for (int k0 = 0; k0 < SEG; k0 += 4) {
        v2f a = *(const v2f*)(Arow2 + k0);
        v2f b;
        b.x = costab[p];
        b.y = costab[(p + t) & (LL - 1)];
        c = __builtin_amdgcn_wmma_f32_16x16x4_f32(
                false, a, false, b, (short)0, c, false, false);
        p = (p + step) & (LL - 1);
    }

#pragma unroll
    for (int r = 0; r < 8; ++r) {
        int m = m0 + r + half * 8;
        out[(size_t)m * LL + t] = c[r];
    }
}

// ---------------------------------------------------------------------------
// Launch. Inputs (setup_inputs order): w_ri(256,64,2) P_ri(1,256,64,2)
// B_ri(1,256,64,2) C_ri(1,256,64,2) log_dt(256) L(=2048, compile-time fixed).
// Output: (1,256,2048) fp32. Workspace: A matrix, 256*2056*4 B ~ 2.1 MB.
// ---------------------------------------------------------------------------
extern "C" void kernel_launch(void* const* d_in, const int* in_sizes, int n_in,
                              void* d_out, int out_size, void* d_ws, size_t ws_size,
                              hipStream_t stream)
{
    const float* w_ri   = (const float*)d_in[0];
    const float* P_ri   = (const float*)d_in[1];
    const float* B_ri   = (const float*)d_in[2];
    const float* C_ri   = (const float*)d_in[3];
    const float* log_dt = (const float*)d_in[4];
    (void)in_sizes; (void)n_in; (void)out_size; (void)ws_size;

    float* Amat = (float*)d_ws;          // HH * KP fp32
    float* out  = (float*)d_out;         // HH * LL fp32

    const int total = HH * LH;
    hipLaunchKernelGGL(hippo_cauchy_kernel,
                       dim3((total + 255) / 256), dim3(256), 0, stream,
                       w_ri, P_ri, B_ri, C_ri, log_dt, Amat);

    // 16 x 16 blocks of 8 waves; each wave computes one 16x16 tile of (256,2048)
    hipLaunchKernelGGL(hippo_idft_wmma_kernel,
                       dim3(LL / (16 * 8), HH / 16), dim3(256), 0, stream,
                       Amat, out);
}